// PermEquivDirGraphVectorField_12206297055728
// MI455X (gfx1250) — compile-verified
//
#include <hip/hip_runtime.h>
#include <hip/hip_bf16.h>

typedef float v2f __attribute__((ext_vector_type(2)));
typedef float v8f __attribute__((ext_vector_type(8)));

#define NN 1024
#define DD 64

// ---------------------------------------------------------------------------
// tg[j] = mean_i t_grad[i, j]
// ---------------------------------------------------------------------------
__global__ __launch_bounds__(256)
void tg_kernel(const float* __restrict__ t_grad, float* __restrict__ tg) {
    int j = blockIdx.x * 256 + threadIdx.x;
    float s = 0.0f;
    for (int i = 0; i < NN; ++i) s += t_grad[(size_t)i * NN + j];
    tg[j] = s * (1.0f / (float)NN);
}

// ---------------------------------------------------------------------------
// rc[((k*2+which)*N + n)*8 + h]:
//   which=0: row[k][n][h] = sum_i idx_emb[n,i] * W1[k][h][1+i]
//   which=1: col[k][n][h] = sum_i idx_emb[n,i] * W1[k][h][65+i]
// ---------------------------------------------------------------------------
__global__ __launch_bounds__(256)
void rowcol_kernel(const float* __restrict__ idx_emb,
                   const float* __restrict__ W1,   // (2, 8, 129)
                   float* __restrict__ rc) {
    int n = blockIdx.x * 256 + threadIdx.x;
    float acc[2][2][8];
#pragma unroll
    for (int k = 0; k < 2; ++k)
#pragma unroll
        for (int w = 0; w < 2; ++w)
#pragma unroll
            for (int h = 0; h < 8; ++h) acc[k][w][h] = 0.0f;

    for (int i = 0; i < 64; ++i) {
        float e = idx_emb[(size_t)n * 64 + i];
#pragma unroll
        for (int k = 0; k < 2; ++k)
#pragma unroll
            for (int h = 0; h < 8; ++h) {
                const float* wrow = W1 + (size_t)(k * 8 + h) * 129;
                acc[k][0][h] = fmaf(e, wrow[1 + i],  acc[k][0][h]);
                acc[k][1][h] = fmaf(e, wrow[65 + i], acc[k][1][h]);
            }
    }
#pragma unroll
    for (int k = 0; k < 2; ++k)
#pragma unroll
        for (int w = 0; w < 2; ++w)
#pragma unroll
            for (int h = 0; h < 8; ++h)
                rc[((size_t)(k * 2 + w) * NN + n) * 8 + h] = acc[k][w][h];
}

// ---------------------------------------------------------------------------
// Fused per-pair message MLP: out[n,m] for both heads (A from adj, AD from
// adj_deriv).  Weights are block-uniform -> scalar loads.
// ---------------------------------------------------------------------------
__global__ __launch_bounds__(256)
void msg_kernel(const float* __restrict__ adj, const float* __restrict__ adj_deriv,
                const float* __restrict__ rc,
                const float* __restrict__ W1, const float* __restrict__ b1,
                const float* __restrict__ W2, const float* __restrict__ b2,
                const float* __restrict__ W3, const float* __restrict__ b3,
                float* __restrict__ Aout, float* __restrict__ ADout) {
    size_t pair = (size_t)blockIdx.x * 256 + threadIdx.x;
    int n = (int)(pair >> 10);
    int m = (int)(pair & (NN - 1));

#pragma unroll
    for (int k = 0; k < 2; ++k) {
        float v = (k == 0) ? adj[pair] : adj_deriv[pair];
        const float* row = rc + ((size_t)(k * 2 + 0) * NN + n) * 8;
        const float* col = rc + ((size_t)(k * 2 + 1) * NN + m) * 8;

        float h1[8];
#pragma unroll
        for (int h = 0; h < 8; ++h) {
            float wa = W1[(size_t)(k * 8 + h) * 129];     // W1[k][h][0]
            h1[h] = fmaxf(fmaf(v, wa, row[h] + col[h] + b1[k * 8 + h]), 0.0f);
        }
        float h2[8];
#pragma unroll
        for (int o = 0; o < 8; ++o) {
            float s = b2[k * 8 + o];
#pragma unroll
            for (int h = 0; h < 8; ++h)
                s = fmaf(W2[(size_t)(k * 8 + o) * 8 + h], h1[h], s);
            h2[o] = fmaxf(s, 0.0f);
        }
        float out = b3[k];
#pragma unroll
        for (int h = 0; h < 8; ++h) out = fmaf(W3[k * 8 + h], h2[h], out);

        if (k == 0) Aout[pair] = out; else ADout[pair] = out;
    }
}

// ---------------------------------------------------------------------------
// T[2*blockIdx.y + TRANSP] = M_y(^T) @ x,  M_0 = A, M_1 = AD,  x: (N, D).
// grid = (32 row-pairs, 2 matrices), block = 128 (4 waves).
// Each wave owns a 32x16 output block (two 16x16 M-tiles sharing one B
// operand -> 2 WMMAs per B fetch).  Branch-free inner loop (TRANSP is a
// template parameter), K marched in steps of 4 via v_wmma_f32_16x16x4_f32.
// ---------------------------------------------------------------------------
template <bool TRANSP>
__global__ __launch_bounds__(128)
void spmm_kernel(const float* __restrict__ A, const float* __restrict__ AD,
                 const float* __restrict__ x, float* __restrict__ T) {
    const int lane = threadIdx.x & 31;
    const int wave = threadIdx.x >> 5;
    const float* M = blockIdx.y ? AD : A;          // uniform, hoisted
    const int m0 = blockIdx.x * 32;                // output row base (2 tiles)
    const int n0 = wave * 16;                      // output col base
    const int half = lane >> 4;                    // 0/1
    const int l16 = lane & 15;

    v8f acc0 = {};
    v8f acc1 = {};
#pragma unroll 2
    for (int k = 0; k < NN; k += 4) {
        v2f avec0, avec1;
        if (!TRANSP) {
            // lane holds M[m0+l16, k+2*half+{0,1}] (contiguous pair)
            const float* p = M + (size_t)(m0 + l16) * NN + (k + 2 * half);
            avec0 = *(const v2f*)p;
            avec1 = *(const v2f*)(p + 16 * NN);
        } else {
            // lane holds M[k+2*half+{0,1}, m0+l16]
            const float* p = M + (size_t)(k + 2 * half) * NN + (m0 + l16);
            avec0.x = p[0];
            avec0.y = p[NN];
            avec1.x = p[16];
            avec1.y = p[NN + 16];
        }
        // B operand: VGPR v holds x[k + 2*half + v, n0+l16]
        const float* q = x + (size_t)(k + 2 * half) * DD + (n0 + l16);
        v2f bvec;
        bvec.x = q[0];
        bvec.y = q[DD];

        acc0 = __builtin_amdgcn_wmma_f32_16x16x4_f32(
            false, avec0, false, bvec, (short)0, acc0, false, false);
        acc1 = __builtin_amdgcn_wmma_f32_16x16x4_f32(
            false, avec1, false, bvec, (short)0, acc1, false, false);
    }

    // C/D layout: VGPR r -> row r (lanes 0-15) / row r+8 (lanes 16-31), col l16
    float* out = T + (size_t)(2 * blockIdx.y + (TRANSP ? 1 : 0)) * (NN * DD);
#pragma unroll
    for (int r = 0; r < 8; ++r) {
        int m = m0 + r + 8 * half;
        out[(size_t)m * DD + (n0 + l16)]        = acc0[r];
        out[(size_t)(m + 16) * DD + (n0 + l16)] = acc1[r];
    }
}

// ---------------------------------------------------------------------------
// xout = T0@W0 + T1@W1 + T2@W2 + T3@W3 + xin@W4 + b, then
//   MODE==0: relu        (layers 0,1)
//   MODE==1: * tg[row]   (final layer)
// grid = 64 row-tiles, block = 128 (4 waves, one 16-col tile each), K=64.
// ---------------------------------------------------------------------------
template <int MODE>
__global__ __launch_bounds__(128)
void combine_kernel(const float* __restrict__ T,   // 4 * N * D
                    const float* __restrict__ xin, // N * D
                    const float* __restrict__ W,   // 5 * D * D (layer slice)
                    const float* __restrict__ b,   // D
                    const float* __restrict__ tg,  // N
                    float* __restrict__ xout) {
    const int lane = threadIdx.x & 31;
    const int wave = threadIdx.x >> 5;
    const int m0 = blockIdx.x * 16;
    const int n0 = wave * 16;
    const int half = lane >> 4;
    const int l16 = lane & 15;

    v8f acc = {};
#pragma unroll 1
    for (int p = 0; p < 5; ++p) {
        const float* src = (p < 4) ? (T + (size_t)p * NN * DD) : xin;
        const float* Wp = W + (size_t)p * DD * DD;
#pragma unroll
        for (int k = 0; k < DD; k += 4) {
            const float* pa = src + (size_t)(m0 + l16) * DD + (k + 2 * half);
            v2f avec = *(const v2f*)pa;
            const float* pb = Wp + (size_t)(k + 2 * half) * DD + (n0 + l16);
            v2f bvec;
            bvec.x = pb[0];
            bvec.y = pb[DD];
            acc = __builtin_amdgcn_wmma_f32_16x16x4_f32(
                false, avec, false, bvec, (short)0, acc, false, false);
        }
    }

#pragma unroll
    for (int r = 0; r < 8; ++r) {
        int m = m0 + r + 8 * half;
        int n = n0 + l16;
        float v = acc[r] + b[n];
        if (MODE == 0) v = fmaxf(v, 0.0f);
        else           v *= tg[m];
        xout[(size_t)m * DD + n] = v;
    }
}

// ---------------------------------------------------------------------------
extern "C" void kernel_launch(void* const* d_in, const int* in_sizes, int n_in,
                              void* d_out, int out_size, void* d_ws, size_t ws_size,
                              hipStream_t stream) {
    const float* y         = (const float*)d_in[0];
    const float* adj       = (const float*)d_in[1];
    const float* adj_deriv = (const float*)d_in[2];
    const float* t_grad    = (const float*)d_in[3];
    const float* idx_emb   = (const float*)d_in[4];
    const float* msg_W1    = (const float*)d_in[5];
    const float* msg_b1    = (const float*)d_in[6];
    const float* msg_W2    = (const float*)d_in[7];
    const float* msg_b2    = (const float*)d_in[8];
    const float* msg_W3    = (const float*)d_in[9];
    const float* msg_b3    = (const float*)d_in[10];
    const float* gnn_W     = (const float*)d_in[11];
    const float* gnn_b     = (const float*)d_in[12];
    float* out = (float*)d_out;

    // workspace carve-up (floats)
    float* ws = (float*)d_ws;
    float* A  = ws;                       // N*N
    float* AD = A  + (size_t)NN * NN;     // N*N
    float* rc = AD + (size_t)NN * NN;     // 4*N*8
    float* tg = rc + (size_t)4 * NN * 8;  // N
    float* T  = tg + NN;                  // 4*N*D
    float* x0 = T  + (size_t)4 * NN * DD; // N*D
    float* x1 = x0 + (size_t)NN * DD;     // N*D

    // 1) column-mean of t_grad
    tg_kernel<<<NN / 256, 256, 0, stream>>>(t_grad, tg);

    // 2) row/col embeddings for both heads
    rowcol_kernel<<<NN / 256, 256, 0, stream>>>(idx_emb, msg_W1, rc);

    // 3) fused message MLP -> A, AD
    msg_kernel<<<(NN * NN) / 256, 256, 0, stream>>>(
        adj, adj_deriv, rc, msg_W1, msg_b1, msg_W2, msg_b2, msg_W3, msg_b3,
        A, AD);

    // 4) three GNN layers (WMMA GEMMs)
    const float* xin = y;
    float* xbuf[2] = {x0, x1};
    for (int l = 0; l < 3; ++l) {
        spmm_kernel<false><<<dim3(NN / 32, 2), 128, 0, stream>>>(A, AD, xin, T);
        spmm_kernel<true ><<<dim3(NN / 32, 2), 128, 0, stream>>>(A, AD, xin, T);
        const float* Wl = gnn_W + (size_t)l * 5 * DD * DD;
        const float* bl = gnn_b + (size_t)l * DD;
        if (l < 2) {
            combine_kernel<0><<<NN / 16, 128, 0, stream>>>(T, xin, Wl, bl, tg, xbuf[l]);
            xin = xbuf[l];
        } else {
            combine_kernel<1><<<NN / 16, 128, 0, stream>>>(T, xin, Wl, bl, tg, out);
        }
    }
}